// CapsRoutingLayer_70884140253958
// MI455X (gfx1250) — compile-verified
//
#include <hip/hip_runtime.h>
#include <math.h>

typedef __attribute__((ext_vector_type(2))) float v2f;
typedef __attribute__((ext_vector_type(8))) float v8f;

#define B_SZ   64
#define NI     2048
#define DI     8
#define NO     32
#define DO_    16
#define NOD    512                 // NO*DO_
#define KTOT   (NI*DI)             // 16384
#define WN     (NO*DO_*DI)         // 4096 floats per-n slice of W

#define NB_GEMM  128               // K-split blocks for iter-0 GEMM
#define KCHUNK   (KTOT/NB_GEMM)    // 128
#define NB_ROUTE 128
#define NPB      (NI/NB_ROUTE)     // 16 capsules per routing block
#define PS_STRIDE (B_SZ*NOD)       // 32768 floats per partial-s block

// ---------------------------------------------------------------------------
// Kernel 1: s0 = sum_{n,i} x[b,n,i] * W[n,o,d,i]   (uniform c folded in later)
// One fp32 GEMM M=64, N=512, K=16384, K-split across blocks, fp32 WMMA 16x16x4.
// Wave w owns o-tiles {w, w+8, w+16, w+24} x all 4 M-tiles.
// ---------------------------------------------------------------------------
__global__ __launch_bounds__(256)
void caps_s0_gemm(const float* __restrict__ x, const float* __restrict__ W,
                  float* __restrict__ ps) {
  const int lane = threadIdx.x & 31;
  const int wave = threadIdx.x >> 5;
  const int half = lane >> 4;          // K-half select per ISA A/B layout
  const int l15  = lane & 15;          // M (for A) / N column (for B, C, D)
  const int kbase = blockIdx.x * KCHUNK;
  float* psb = ps + (size_t)blockIdx.x * PS_STRIDE;

  for (int ot = 0; ot < 4; ++ot) {
    const int o = wave + ot * 8;
    const float* wc = W + (size_t)(o * DO_ + l15) * DI;   // column (o,d=l15)
    for (int mt = 0; mt < 4; ++mt) {
      const int row = mt * 16 + l15;
      const float* xr = x + (size_t)row * KTOT;
      v8f acc = {};
      for (int k0 = kbase; k0 < kbase + KCHUNK; k0 += 4) {
        const int ka = k0 + 2 * half;          // VGPR0/1 hold K=ka, ka+1
        v2f a  = *(const v2f*)(xr + ka);
        const int n = ka >> 3, i = ka & 7;     // i,i+1 stay in same n (k0%4==0)
        v2f bv = *(const v2f*)(wc + (size_t)n * WN + i);
        acc = __builtin_amdgcn_wmma_f32_16x16x4_f32(
                  false, a, false, bv, (short)0, acc, false, false);
      }
#pragma unroll
      for (int r = 0; r < 8; ++r) {            // D vgpr r -> b row r (+8 hi half)
        const int b = mt * 16 + r + 8 * half;
        psb[b * NOD + o * DO_ + l15] = acc[r];
      }
    }
  }
}

// ---------------------------------------------------------------------------
// Routing pass (iterations 1 and 2). Block owns NPB capsules n; per n:
//  phase1: WMMA-recompute x_hat tiles, logits r[b,o] = <v, x_hat[b,o,:]>
//          (lane butterfly over d), iter1 stores logits, iter2 adds stored.
//  phase2: softmax over o per b in LDS.
//  phase3: WMMA-recompute tiles, s[b,o,d] += c[b,o]*x_hat — accumulated in
//          per-lane registers (owner-computes, deterministic), flushed once.
// ---------------------------------------------------------------------------
__global__ __launch_bounds__(256)
void caps_route(const float* __restrict__ x, const float* __restrict__ W,
                const float* __restrict__ vcur, float* __restrict__ rlog,
                float* __restrict__ ps, int iter) {
  __shared__ float r1s[B_SZ * NO];   // logits for current n (8 KB)
  __shared__ float cs[B_SZ * NO];    // coupling coeffs     (8 KB)
  const int lane = threadIdx.x & 31;
  const int wave = threadIdx.x >> 5;
  const int half = lane >> 4;
  const int l15  = lane & 15;

  v8f zero = {};
  v8f sacc[16];
#pragma unroll
  for (int t = 0; t < 16; ++t) sacc[t] = zero;

  const int n0 = blockIdx.x * NPB;
  for (int nn = 0; nn < NPB; ++nn) {
    const int n = n0 + nn;
    const float* wn = W + (size_t)n * WN;

    // ---- phase 1: agreement logits
    for (int ot = 0; ot < 4; ++ot) {
      const int o = wave + ot * 8;
      for (int mt = 0; mt < 4; ++mt) {
        const int row = mt * 16 + l15;
        v8f d = zero;
#pragma unroll
        for (int ks = 0; ks < 2; ++ks) {       // K=8 -> two k-steps of 4
          const int k = ks * 4 + 2 * half;
          v2f a  = *(const v2f*)(x + (size_t)row * KTOT + n * DI + k);
          v2f bv = *(const v2f*)(wn + (size_t)(o * DO_ + l15) * DI + k);
          d = __builtin_amdgcn_wmma_f32_16x16x4_f32(
                  false, a, false, bv, (short)0, d, false, false);
        }
#pragma unroll
        for (int r = 0; r < 8; ++r) {
          const int b = mt * 16 + r + 8 * half;
          float p = d[r] * vcur[b * NOD + o * DO_ + l15];
          p += __shfl_xor(p, 1);
          p += __shfl_xor(p, 2);
          p += __shfl_xor(p, 4);
          p += __shfl_xor(p, 8);               // sum over d within 16-lane half
          if (l15 == 0) {
            const size_t ridx = ((size_t)b * NI + n) * NO + o;
            float logit;
            if (iter == 1) { logit = p; rlog[ridx] = p; }
            else           { logit = rlog[ridx] + p; }
            r1s[b * NO + o] = logit;
          }
        }
      }
    }
    __syncthreads();

    // ---- phase 2: softmax over output capsules
    if (threadIdx.x < B_SZ) {
      const int b = threadIdx.x;
      float m = -3.4e38f;
      for (int o = 0; o < NO; ++o) m = fmaxf(m, r1s[b * NO + o]);
      float ssum = 0.f;
      for (int o = 0; o < NO; ++o) {
        float e = __expf(r1s[b * NO + o] - m);
        cs[b * NO + o] = e; ssum += e;
      }
      const float inv = 1.f / ssum;
      for (int o = 0; o < NO; ++o) cs[b * NO + o] *= inv;
    }
    __syncthreads();

    // ---- phase 3: s += c * x_hat (recompute tiles, accumulate in registers)
    for (int ot = 0; ot < 4; ++ot) {
      const int o = wave + ot * 8;
      for (int mt = 0; mt < 4; ++mt) {
        const int row = mt * 16 + l15;
        v8f d = zero;
#pragma unroll
        for (int ks = 0; ks < 2; ++ks) {
          const int k = ks * 4 + 2 * half;
          v2f a  = *(const v2f*)(x + (size_t)row * KTOT + n * DI + k);
          v2f bv = *(const v2f*)(wn + (size_t)(o * DO_ + l15) * DI + k);
          d = __builtin_amdgcn_wmma_f32_16x16x4_f32(
                  false, a, false, bv, (short)0, d, false, false);
        }
#pragma unroll
        for (int r = 0; r < 8; ++r) {
          const int b = mt * 16 + r + 8 * half;
          sacc[ot * 4 + mt][r] += cs[b * NO + o] * d[r];
        }
      }
    }
    __syncthreads();
  }

  // ---- flush per-block partial s
  float* psb = ps + (size_t)blockIdx.x * PS_STRIDE;
  for (int ot = 0; ot < 4; ++ot) {
    const int o = wave + ot * 8;
    for (int mt = 0; mt < 4; ++mt) {
      v8f s = sacc[ot * 4 + mt];
#pragma unroll
      for (int r = 0; r < 8; ++r) {
        const int b = mt * 16 + r + 8 * half;
        psb[b * NOD + o * DO_ + l15] = s[r];
      }
    }
  }
}

// ---------------------------------------------------------------------------
// Reduce partial s over blocks, apply optional prescale (1/No for iter 0),
// squash per (b,o): v = (||s||/(1+||s||^2)) * s. One block per batch element.
// ---------------------------------------------------------------------------
__global__ __launch_bounds__(512)
void caps_reduce_squash(const float* __restrict__ ps, int nparts, float prescale,
                        float* __restrict__ out) {
  __shared__ float sh[NOD];
  __shared__ float scl[NO];
  const int b  = blockIdx.x;
  const int od = threadIdx.x;
  float s = 0.f;
  for (int p = 0; p < nparts; ++p)
    s += ps[(size_t)p * PS_STRIDE + b * NOD + od];
  s *= prescale;
  sh[od] = s;
  __syncthreads();
  if (od < NO) {
    float nn = 0.f;
#pragma unroll
    for (int d = 0; d < DO_; ++d) { float t = sh[od * DO_ + d]; nn += t * t; }
    scl[od] = sqrtf(nn) / (1.f + nn);
  }
  __syncthreads();
  out[(size_t)b * NOD + od] = sh[od] * scl[od >> 4];
}

// ---------------------------------------------------------------------------
extern "C" void kernel_launch(void* const* d_in, const int* in_sizes, int n_in,
                              void* d_out, int out_size, void* d_ws, size_t ws_size,
                              hipStream_t stream) {
  const float* x = (const float*)d_in[0];   // (64, 2048, 8)
  const float* W = (const float*)d_in[1];   // (2048, 32, 16, 8)
  float* out = (float*)d_out;               // (64, 32, 16)

  float* ps   = (float*)d_ws;                          // 128 * 32768 floats (16 MB)
  float* vbuf = ps + (size_t)NB_GEMM * PS_STRIDE;      // 32768 floats
  float* rlog = vbuf + (size_t)B_SZ * NOD;             // 64*2048*32 floats (16 MB)

  // iter 0: uniform coupling 1/32 -> single big fp32 WMMA GEMM + squash
  caps_s0_gemm<<<NB_GEMM, 256, 0, stream>>>(x, W, ps);
  caps_reduce_squash<<<B_SZ, NOD, 0, stream>>>(ps, NB_GEMM, 1.0f / NO, vbuf);
  // iter 1: logits from v0, softmax, weighted sum, squash
  caps_route<<<NB_ROUTE, 256, 0, stream>>>(x, W, vbuf, rlog, ps, 1);
  caps_reduce_squash<<<B_SZ, NOD, 0, stream>>>(ps, NB_ROUTE, 1.0f, vbuf);
  // iter 2: logits += agreement with v1, softmax, weighted sum, squash -> out
  caps_route<<<NB_ROUTE, 256, 0, stream>>>(x, W, vbuf, rlog, ps, 2);
  caps_reduce_squash<<<B_SZ, NOD, 0, stream>>>(ps, NB_ROUTE, 1.0f, out);
}